// LUT_unused_80032420594223
// MI455X (gfx1250) — compile-verified
//
#include <hip/hip_runtime.h>
#include <stdint.h>
#include <stddef.h>

// ---------------------------------------------------------------------------
// 4D tetrahedral LUT interpolation + 4x nearest-neighbor upscale (MI455X).
// Memory-bound streaming kernel: 144 B/pixel, ~453 MB total -> ~19.5 us at
// 23.3 TB/s. Uses CDNA5 async global->LDS DMA (ASYNCcnt) to stage the 20
// input streams, NT temporal hints for the one-pass stream, B128 accesses.
// ---------------------------------------------------------------------------

typedef float v4f __attribute__((ext_vector_type(4)));
typedef int   v4i __attribute__((ext_vector_type(4)));

#define GLOBAL_AS __attribute__((address_space(1)))
#define LDS_AS    __attribute__((address_space(3)))

typedef GLOBAL_AS v4i* gv4i_p;   // global (AS1) pointer to b128 payload
typedef LDS_AS    v4i* lv4i_p;   // LDS (AS3) pointer to b128 payload

#if defined(__has_builtin)
#  if __has_builtin(__builtin_amdgcn_global_load_async_to_lds_b128)
#    define ASYNC_OK 1
#  endif
#  if __has_builtin(__builtin_amdgcn_s_wait_asynccnt)
#    define HAVE_WAIT_ASYNC 1
#  endif
#endif
#ifndef ASYNC_OK
#  define ASYNC_OK 0
#endif
#ifndef HAVE_WAIT_ASYNC
#  define HAVE_WAIT_ASYNC 0
#endif

// SHAPE = (4,3,512,512), Q = 16, UPSCALE = 4 (fixed by the reference)
#define W_MASK  511
#define W_SHIFT 9
#define ROW_F   2048     // floats per upscaled output row (W*UP)
#define PLANE_F 8192     // floats per (b,c,h) plane (UP*W*UP)

struct Args {
  const float* p[16];   // p0000 .. p1111 (binary order, d = LSB)
  const int*   f[4];    // fa, fb, fc, fd
  float*       out;
  int          n4;      // number of 4-pixel groups
};

// One pixel: branchless rank-sort of (fa,fb,fc,fd) + 5-corner blend.
__device__ __forceinline__ float lut1(const float* ps, int fa, int fb, int fc, int fd) {
  const int ab = fa > fb, ac = fa > fc, ad = fa > fd;
  const int bc = fb > fc, bd = fb > fd, cd = fc > fd;
  // Ranks under a consistent tie-break ("later letter wins" on equality).
  // Ties give identical blends since the differing coefficient is exactly 0.
  const int ra = ab + ac + ad;
  const int rb = (1 - ab) + bc + bd;
  const int rc = (2 - ac - bc) + cd;
  const float A = (float)fa, B = (float)fb, C = (float)fc, D = (float)fd;
  const float v1 = (ra == 3) ? A : (rb == 3) ? B : (rc == 3) ? C : D;
  const float v2 = (ra == 2) ? A : (rb == 2) ? B : (rc == 2) ? C : D;
  const float v3 = (ra == 1) ? A : (rb == 1) ? B : (rc == 1) ? C : D;
  const float v4 = (A + B + C + D) - v1 - v2 - v3;   // exact: small integers
  // corner after adding the largest coordinate's bit (a=8,b=4,c=2,d=1)
  const float pA = (ra == 3) ? ps[8] : (rb == 3) ? ps[4] : (rc == 3) ? ps[2] : ps[1];
  // corner with all bits except the smallest coordinate's bit
  const float pC = (ra == 0) ? ps[7] : (rb == 0) ? ps[11] : (rc == 0) ? ps[13] : ps[14];
  // corner with the top-two coordinates' bits
  const bool a2 = ra >= 2, b2 = rb >= 2, c2 = rc >= 2;
  const float pB = (a2 && b2) ? ps[12]
                 : (a2 && c2) ? ps[10]
                 : a2         ? ps[9]
                 : (b2 && c2) ? ps[6]
                 : b2         ? ps[5]
                 :              ps[3];
  const float r = (16.0f - v1) * ps[0] + (v1 - v2) * pA + (v2 - v3) * pB +
                  (v3 - v4) * pC + v4 * ps[15];
  return r * 0.0625f;  // divide by Q
}

// Compute 4 pixels and emit the 4x4 pixel-shuffle upscale with NT B128 stores.
__device__ __forceinline__ void body(float* __restrict__ out, int g4,
                                     const v4f* pv, const v4i* fv) {
  float r[4];
#pragma unroll
  for (int j = 0; j < 4; ++j) {
    float ps[16];
#pragma unroll
    for (int k = 0; k < 16; ++k) ps[k] = pv[k][j];
    r[j] = lut1(ps, fv[0][j], fv[1][j], fv[2][j], fv[3][j]);
  }
  const int g  = g4 << 2;          // first pixel id
  const int w0 = g & W_MASK;       // aligned to 4
  const int P  = g >> W_SHIFT;     // (b,c,h) plane index
  float* ob = out + (size_t)P * PLANE_F + (size_t)w0 * 4;
#pragma unroll
  for (int i = 0; i < 4; ++i) {    // upscale rows
#pragma unroll
    for (int j = 0; j < 4; ++j) {  // 4 source pixels, 4 replicas each
      v4f q = {r[j], r[j], r[j], r[j]};
      __builtin_nontemporal_store(q, (v4f*)(ob + i * ROW_F + j * 4));
    }
  }
}

#if ASYNC_OK
// Async path: DMA the 20 input streams into LDS (ASYNCcnt), wait, read back
// with ds_load_b128. 40 KB LDS per 128-thread block -> 8 blocks/WGP.
__global__ __launch_bounds__(128) void lut4d_async(Args a) {
  __shared__ v4f sbuf[20][128];
  const int t  = threadIdx.x;
  const int g4 = blockIdx.x * 128 + t;
  if (g4 >= a.n4) return;
#pragma unroll
  for (int k = 0; k < 16; ++k)
    __builtin_amdgcn_global_load_async_to_lds_b128(
        (gv4i_p)((const v4f*)a.p[k] + g4),
        (lv4i_p)&sbuf[k][t], 0, 0);
#pragma unroll
  for (int k = 0; k < 4; ++k)
    __builtin_amdgcn_global_load_async_to_lds_b128(
        (gv4i_p)((const v4i*)a.f[k] + g4),
        (lv4i_p)&sbuf[16 + k][t], 0, 0);
#if HAVE_WAIT_ASYNC
  __builtin_amdgcn_s_wait_asynccnt(0);
  asm volatile("" ::: "memory");   // keep LDS readbacks below the wait
#else
  asm volatile("s_wait_asynccnt 0" ::: "memory");
#endif
  v4f pv[16];
#pragma unroll
  for (int k = 0; k < 16; ++k) pv[k] = sbuf[k][t];
  v4i fv[4];
#pragma unroll
  for (int k = 0; k < 4; ++k) fv[k] = *(const v4i*)&sbuf[16 + k][t];
  body(a.out, g4, pv, fv);
}
#endif

// Fallback / reference path: direct NT B128 loads.
__global__ __launch_bounds__(256) void lut4d_direct(Args a) {
  const int g4 = blockIdx.x * 256 + threadIdx.x;
  if (g4 >= a.n4) return;
  v4f pv[16];
#pragma unroll
  for (int k = 0; k < 16; ++k)
    pv[k] = __builtin_nontemporal_load((const v4f*)a.p[k] + g4);
  v4i fv[4];
#pragma unroll
  for (int k = 0; k < 4; ++k)
    fv[k] = __builtin_nontemporal_load((const v4i*)a.f[k] + g4);
  body(a.out, g4, pv, fv);
}

extern "C" void kernel_launch(void* const* d_in, const int* in_sizes, int n_in,
                              void* d_out, int out_size, void* d_ws, size_t ws_size,
                              hipStream_t stream) {
  (void)n_in; (void)out_size; (void)d_ws; (void)ws_size;
  Args a;
  for (int k = 0; k < 16; ++k) a.p[k] = (const float*)d_in[k];
  for (int k = 0; k < 4;  ++k) a.f[k] = (const int*)d_in[16 + k];
  a.out = (float*)d_out;
  const int n = in_sizes[0];          // 4*3*512*512 = 3,145,728 pixels
  a.n4 = n >> 2;                      // 786,432 four-pixel groups
#if ASYNC_OK
  const int block = 128;
  const int grid  = (a.n4 + block - 1) / block;
  hipLaunchKernelGGL(lut4d_async, dim3(grid), dim3(block), 0, stream, a);
#else
  const int block = 256;
  const int grid  = (a.n4 + block - 1) / block;
  hipLaunchKernelGGL(lut4d_direct, dim3(grid), dim3(block), 0, stream, a);
#endif
}